// PINN_10213432229975
// MI455X (gfx1250) — compile-verified
//
#include <hip/hip_runtime.h>
#include <hip/hip_bf16.h>
#include <math.h>

typedef float v2f __attribute__((ext_vector_type(2)));
typedef float v8f __attribute__((ext_vector_type(8)));

#if defined(__HIP_DEVICE_COMPILE__) && !__has_builtin(__builtin_amdgcn_wmma_f32_16x16x4_f32)
#error "wmma f32 16x16x4 builtin not available on this toolchain"
#endif

#define N_NODES 100000
#define N_EDGES 1000000
#define DX 8
#define HD 128
#define N_DOM 32768
#define N_BND 8192
#define N_SP (N_DOM + N_BND)
#define INV_RE 1.45e-05f

// d_out layout (floats): out_sup[300000] | continuity[32768] | momx[32768] | momy[32768] | boundary[8192]
#define OFF_CONT 300000
#define OFF_MOMX 332768
#define OFF_MOMY 365536
#define OFF_BND  398304

#define ASTRIDE 130   // padded LDS row stride for layer-1 panel (even -> 8B aligned v2f loads)
#define HSTRIDE 129   // padded LDS row stride for h2 panel

// ---------------- kernel 1: node_h = tanh(x@Wx + pos2@Wp) ----------------
__global__ void node_h_kernel(const float* __restrict__ x,
                              const float* __restrict__ pos,
                              const float* __restrict__ Wx,
                              const float* __restrict__ Wp,
                              float* __restrict__ node_h) {
  int t = blockIdx.x * blockDim.x + threadIdx.x;
  if (t >= N_NODES * HD) return;
  int n = t >> 7, j = t & 127;
  const float* xr = x + n * DX;
  float acc = 0.f;
#pragma unroll
  for (int k = 0; k < DX; ++k) acc = fmaf(xr[k], Wx[k * HD + j], acc);
  acc = fmaf(pos[n * 3 + 0], Wp[0 * HD + j], acc);
  acc = fmaf(pos[n * 3 + 1], Wp[1 * HD + j], acc);
  node_h[t] = tanhf(acc);
}

// ---------------- kernel 2: edge messages + atomic segment_sum ----------------
// one wave32 per edge; each lane handles 4 channels
__global__ void edge_kernel(const int* __restrict__ ei,
                            const float* __restrict__ eattr,
                            const float* __restrict__ We,
                            const float* __restrict__ node_h,
                            float* __restrict__ agg) {
  int wid = (blockIdx.x * blockDim.x + threadIdx.x) >> 5;
  int lane = threadIdx.x & 31;
  if (wid >= N_EDGES) return;
  int src = ei[wid];
  int dst = ei[N_EDGES + wid];
  float a0 = eattr[wid * 4 + 0];
  float a1 = eattr[wid * 4 + 1];
  float a3 = eattr[wid * 4 + 3];   // reference selects edge_attr[:, [0,1,3]]
  const float* nh = node_h + (long long)src * HD;
  float* ag = agg + (long long)dst * HD;
#pragma unroll
  for (int q = 0; q < 4; ++q) {
    int c = lane + q * 32;
    float w = a0 * We[0 * HD + c] + a1 * We[1 * HD + c] + a3 * We[2 * HD + c];
    atomicAdd(&ag[c], nh[c] * w);
  }
}

// ---------------- kernel 3: h = tanh(node_h + agg) (in place) + ctx partial sums ----------------
__global__ void hctx_kernel(const float* __restrict__ agg,
                            float* __restrict__ node_h,   // becomes h
                            float* __restrict__ ctxsum) {
  int t0 = blockIdx.x * blockDim.x + threadIdx.x;
  int stride = gridDim.x * blockDim.x;    // multiple of 128 -> column fixed per thread
  float local = 0.f;
  for (int t = t0; t < N_NODES * HD; t += stride) {
    float h = tanhf(node_h[t] + agg[t]);
    node_h[t] = h;
    local += h;
  }
  atomicAdd(&ctxsum[t0 & 127], local);
}

// ---------------- kernel 4: out_sup = h @ Wout (128x3), one wave per node ----------------
__global__ void outsup_kernel(const float* __restrict__ h,
                              const float* __restrict__ Wout,
                              float* __restrict__ out) {
  int wid = (blockIdx.x * blockDim.x + threadIdx.x) >> 5;
  int lane = threadIdx.x & 31;
  if (wid >= N_NODES) return;
  const float* hr = h + (long long)wid * HD;
  float acc0 = 0.f, acc1 = 0.f, acc2 = 0.f;
#pragma unroll
  for (int q = 0; q < 4; ++q) {
    int k = lane + 32 * q;
    float hv = hr[k];
    acc0 = fmaf(hv, Wout[k * 3 + 0], acc0);
    acc1 = fmaf(hv, Wout[k * 3 + 1], acc1);
    acc2 = fmaf(hv, Wout[k * 3 + 2], acc2);
  }
#pragma unroll
  for (int m = 16; m >= 1; m >>= 1) {
    acc0 += __shfl_xor(acc0, m, 32);
    acc1 += __shfl_xor(acc1, m, 32);
    acc2 += __shfl_xor(acc2, m, 32);
  }
  if (lane == 0) {
    out[wid * 3 + 0] = acc0;
    out[wid * 3 + 1] = acc1;
    out[wid * 3 + 2] = acc2;
  }
}

// ---------------- kernel 5: prep — cb1 = b1 + (ctxsum/N)@W1[2:], repack W2 into B fragments ----------------
__global__ void prep_kernel(const float* __restrict__ ctxsum,
                            const float* __restrict__ W1,
                            const float* __restrict__ b1,
                            const float* __restrict__ W2,
                            float* __restrict__ cb1,
                            float* __restrict__ Bpack) {
  int tid = threadIdx.x;   // 128 threads, 1 block
  {
    float acc = b1[tid];
    const float inv_n = 1.f / (float)N_NODES;
    for (int k = 0; k < HD; ++k)
      acc = fmaf(ctxsum[k] * inv_n, W1[(2 + k) * HD + tid], acc);
    cb1[tid] = acc;
  }
  // B fragment layout for v_wmma_f32_16x16x4_f32:
  // B is 4x16 (KxN). VGPR0: lanes0-15 = row k0 (n0..n0+15), lanes16-31 = row k0+2; VGPR1: k0+1 / k0+3
  for (int idx = tid; idx < 8 * 32 * 32; idx += 128) {
    int l  = idx & 31;
    int kt = (idx >> 5) & 31;
    int nt = idx >> 10;
    int k0 = 4 * kt + ((l >= 16) ? 2 : 0);
    int n  = nt * 16 + (l & 15);
    Bpack[idx * 2 + 0] = W2[k0 * HD + n];
    Bpack[idx * 2 + 1] = W2[(k0 + 1) * HD + n];
  }
}

// ---------------- kernel 6: PINN sampling — fwd value + 1st/2nd derivs via f32 WMMA ----------------
// block = 128 threads (4 waves), 16 points/block; 5 channels (val,dx,dy,dxx,dyy) x 16 points = 80 A rows
__global__ void __launch_bounds__(128)
pinn_kernel(const float* __restrict__ dom, const float* __restrict__ bnd,
            const float* __restrict__ W1, const float* __restrict__ cb1,
            const float* __restrict__ Bpack, const float* __restrict__ b2,
            const float* __restrict__ W3,
            const float* __restrict__ x, const float* __restrict__ x_mask,
            const int* __restrict__ idxs,
            float* __restrict__ out) {
  __shared__ __align__(16) float Abuf[80 * ASTRIDE];
  __shared__ __align__(16) float hbuf[80 * HSTRIDE];
  __shared__ float splds[32];
  __shared__ float W3lds[HD * 3];
  __shared__ float b2lds[HD];
  __shared__ float reslds[5 * 16 * 3];

  int tid = threadIdx.x;
  int p0 = blockIdx.x * 16;

  if (tid < 32) {
    int gi = p0 * 2 + tid;          // flat index into concat(dom,bnd) viewed (N_SP,2)
    splds[tid] = (gi < N_DOM * 2) ? dom[gi] : bnd[gi - N_DOM * 2];
  }
  for (int i = tid; i < HD * 3; i += 128) W3lds[i] = W3[i];
  if (tid < HD) b2lds[tid] = b2[tid];
  __syncthreads();

  // phase 1: layer-1 value + analytic 1st/2nd derivs wrt (x,y). pre1 is affine in sp.
  {
    int col = tid;
    float w0 = W1[col];             // W1[0][col]
    float w1 = W1[HD + col];        // W1[1][col]
    float cb = cb1[col];            // b1 + ctx@W1[2:]
    for (int p = 0; p < 16; ++p) {
      float sx = splds[p * 2 + 0], sy = splds[p * 2 + 1];
      float a  = tanhf(fmaf(sx, w0, fmaf(sy, w1, cb)));
      float s  = 1.f - a * a;
      float m2 = -2.f * a * s;
      Abuf[(0 * 16 + p) * ASTRIDE + col] = a;
      Abuf[(1 * 16 + p) * ASTRIDE + col] = s * w0;
      Abuf[(2 * 16 + p) * ASTRIDE + col] = s * w1;
      Abuf[(3 * 16 + p) * ASTRIDE + col] = m2 * w0 * w0;
      Abuf[(4 * 16 + p) * ASTRIDE + col] = m2 * w1 * w1;
    }
  }
  __syncthreads();

  // phase 2: (80 x 128) @ W2(128x128) with v_wmma_f32_16x16x4_f32; wave owns 2 N-tiles x 5 channels
  {
    int wave = tid >> 5, lane = tid & 31;
    int half = lane >> 4, lr = lane & 15;
    int abase = lr * ASTRIDE + 2 * half;   // A frag: row = c*16+lr, k = 4*kt + 2*half (+0/+1 in v2f)
    for (int nti = 0; nti < 2; ++nti) {
      int nt = wave * 2 + nti;
      v8f C0 = {}, C1 = {}, C2 = {}, C3 = {}, C4 = {};
      const v2f* Bp = (const v2f*)Bpack + (nt * 32) * 32 + lane;
      for (int kt = 0; kt < 32; ++kt) {
        v2f b = Bp[kt * 32];
        int ao = abase + 4 * kt;
        v2f a0 = *(const v2f*)&Abuf[(0 * 16) * ASTRIDE + ao];
        v2f a1 = *(const v2f*)&Abuf[(1 * 16) * ASTRIDE + ao];
        v2f a2 = *(const v2f*)&Abuf[(2 * 16) * ASTRIDE + ao];
        v2f a3 = *(const v2f*)&Abuf[(3 * 16) * ASTRIDE + ao];
        v2f a4 = *(const v2f*)&Abuf[(4 * 16) * ASTRIDE + ao];
        C0 = __builtin_amdgcn_wmma_f32_16x16x4_f32(false, a0, false, b, (short)0, C0, false, false);
        C1 = __builtin_amdgcn_wmma_f32_16x16x4_f32(false, a1, false, b, (short)0, C1, false, false);
        C2 = __builtin_amdgcn_wmma_f32_16x16x4_f32(false, a2, false, b, (short)0, C2, false, false);
        C3 = __builtin_amdgcn_wmma_f32_16x16x4_f32(false, a3, false, b, (short)0, C3, false, false);
        C4 = __builtin_amdgcn_wmma_f32_16x16x4_f32(false, a4, false, b, (short)0, C4, false, false);
      }
      // tanh chain rule: elementwise across the 5 C tiles (same lane/VGPR = same (point,col))
      int colg = nt * 16 + lr;
      float b2v = b2lds[colg];
#pragma unroll
      for (int v = 0; v < 8; ++v) {
        int p = v + 8 * half;       // C layout: VGPR v, lanes0-15 -> row v, lanes16-31 -> row v+8
        float t  = tanhf(C0[v] + b2v);
        float s  = 1.f - t * t;
        float px = C1[v], py = C2[v];
        hbuf[(0 * 16 + p) * HSTRIDE + colg] = t;
        hbuf[(1 * 16 + p) * HSTRIDE + colg] = s * px;
        hbuf[(2 * 16 + p) * HSTRIDE + colg] = s * py;
        hbuf[(3 * 16 + p) * HSTRIDE + colg] = s * C3[v] - 2.f * t * s * px * px;
        hbuf[(4 * 16 + p) * HSTRIDE + colg] = s * C4[v] - 2.f * t * s * py * py;
      }
    }
  }
  __syncthreads();

  // phase 3a: layer 3 (128 -> 3) for 5 channels x 16 points
  for (int job = tid; job < 240; job += 128) {
    int j = job % 3;
    int p = (job / 3) & 15;
    int c = job / 48;
    const float* hr = &hbuf[(c * 16 + p) * HSTRIDE];
    float acc = 0.f;
    for (int k = 0; k < HD; ++k) acc = fmaf(hr[k], W3lds[k * 3 + j], acc);
    reslds[job] = acc;   // reslds[(c*16+p)*3 + j]
  }
  __syncthreads();

  // phase 3b: residual assembly, one thread per point
  if (tid < 16) {
    int p = tid, gi = p0 + p;
    float u   = reslds[(0*16+p)*3+0], v_  = reslds[(0*16+p)*3+1], pv = reslds[(0*16+p)*3+2];
    float ux  = reslds[(1*16+p)*3+0], vx  = reslds[(1*16+p)*3+1], px = reslds[(1*16+p)*3+2];
    float uy  = reslds[(2*16+p)*3+0], vy  = reslds[(2*16+p)*3+1], py = reslds[(2*16+p)*3+2];
    float uxx = reslds[(3*16+p)*3+0], vxx = reslds[(3*16+p)*3+1];
    float uyy = reslds[(4*16+p)*3+0], vyy = reslds[(4*16+p)*3+1];
    if (gi < N_DOM) {
      out[OFF_CONT + gi] = ux + vy;
      out[OFF_MOMX + gi] = u * ux + v_ * uy + px - (uxx + uyy) * INV_RE;
      out[OFF_MOMY + gi] = u * vx + v_ * vy + py - (vxx + vyy) * INV_RE;
    } else {
      int jb = gi - N_DOM;
      int id = idxs[jb];
      const float* xb = x + (long long)id * DX;
      const float* m  = x_mask + (long long)id * 6;
      float tx = xb[0], ty = xb[1];
      float nx = -ty, ny = tx;
      float r = fabsf(tx * u  + ty * v_ - xb[2]) * m[1]
              + fabsf(nx * u  + ny * v_ - xb[3]) * m[2]
              + fabsf(pv - xb[4]) * m[3]
              + fabsf(nx * ux + ny * vy - xb[5]) * m[4]
              + fabsf(nx * px + ny * py - xb[6]) * m[5];
      float cden = m[1] + m[2] + m[3] + m[4] + m[5];
      out[OFF_BND + jb] = r / cden;
    }
  }
}

extern "C" void kernel_launch(void* const* d_in, const int* in_sizes, int n_in,
                              void* d_out, int out_size, void* d_ws, size_t ws_size,
                              hipStream_t stream) {
  const float* x     = (const float*)d_in[0];
  const float* xmask = (const float*)d_in[1];
  const int*   ei    = (const int*)d_in[2];
  const float* eattr = (const float*)d_in[3];
  const float* pos   = (const float*)d_in[4];
  /* d_in[5] = batch, unused */
  const float* dom   = (const float*)d_in[6];
  const float* bnd   = (const float*)d_in[7];
  const int*   idxs  = (const int*)d_in[8];
  const float* Wx    = (const float*)d_in[9];
  const float* Wp    = (const float*)d_in[10];
  const float* We    = (const float*)d_in[11];
  const float* Wout  = (const float*)d_in[12];
  const float* W1    = (const float*)d_in[13];
  const float* b1    = (const float*)d_in[14];
  const float* W2    = (const float*)d_in[15];
  const float* b2    = (const float*)d_in[16];
  const float* W3    = (const float*)d_in[17];
  float* out = (float*)d_out;

  char* ws = (char*)d_ws;
  float* node_h = (float*)(ws);                              // 51.2 MB (becomes h)
  float* agg    = (float*)(ws + 51200000);                   // 51.2 MB
  float* ctxsum = (float*)(ws + 102400000);                  // 512 B
  float* cb1    = (float*)(ws + 102400512);                  // 512 B
  float* Bpack  = (float*)(ws + 102401024);                  // 64 KB

  (void)hipMemsetAsync(agg, 0, (size_t)N_NODES * HD * sizeof(float), stream);
  (void)hipMemsetAsync(ctxsum, 0, HD * sizeof(float), stream);

  node_h_kernel<<<(N_NODES * HD + 255) / 256, 256, 0, stream>>>(x, pos, Wx, Wp, node_h);
  edge_kernel<<<N_EDGES / 8, 256, 0, stream>>>(ei, eattr, We, node_h, agg);
  hctx_kernel<<<1024, 128, 0, stream>>>(agg, node_h, ctxsum);
  outsup_kernel<<<(N_NODES + 7) / 8, 256, 0, stream>>>(node_h, Wout, out);
  prep_kernel<<<1, 128, 0, stream>>>(ctxsum, W1, b1, W2, cb1, Bpack);
  pinn_kernel<<<N_SP / 16, 128, 0, stream>>>(dom, bnd, W1, cb1, Bpack, b2, W3,
                                             x, xmask, idxs, out);
}